// BatchNorm_58076547777199
// MI455X (gfx1250) — compile-verified
//
#include <hip/hip_runtime.h>
#include <hip/hip_bf16.h>
#include <stdint.h>

#define DIM   480      // 128*1 + 64*3 + 32*5
#define DIM4  120      // float4 per row
#define NCH   224      // 128 + 64 + 32
#define NG    64
#define NEPS  1e-5f

typedef float v4f __attribute__((ext_vector_type(4)));

// ---------------- workspace layout (floats) ----------------
// [0,64)            counts
// [64,8256)         scalar sums        (NG*128)
// [8256,22592)      sum of squares     (NG*224)
// [22592,36928)     scale table        (NG*224)
// [36928,51264)     shift table        (NG*224)
#define WS_CNT    0
#define WS_SUM    64
#define WS_SSQ    8256
#define WS_SCALE  22592
#define WS_SHIFT  36928
#define WS_ZERO_N 22592   // only the atomic accumulators need zeroing

__global__ void zero_stats(float* __restrict__ ws, int n) {
    for (int i = blockIdx.x * blockDim.x + threadIdx.x; i < n;
         i += gridDim.x * blockDim.x)
        ws[i] = 0.0f;
}

// ---------------- pass 1: per-graph statistics ----------------
// block = 128 threads (4 wave32); lanes 0..119 each own 4 consecutive columns
// (one float4 / row -> global_load_b128). Contiguous row ranges per block
// exploit sorted batch_id: register accumulators, atomics only at graph
// boundaries / range end.
__global__ __launch_bounds__(128) void pass1_stats(
    const float* __restrict__ x, const int* __restrict__ seg,
    float* __restrict__ ws, int nrows, int rows_per_block)
{
    __shared__ float lsq[DIM];
    float* cnt  = ws + WS_CNT;
    float* sums = ws + WS_SUM;
    float* ssq  = ws + WS_SSQ;

    const int  t      = threadIdx.x;
    const bool active = t < DIM4;
    const int  r0 = blockIdx.x * rows_per_block;
    int r1 = r0 + rows_per_block;
    if (r1 > nrows) r1 = nrows;
    if (r0 >= r1) return;

    const float4* __restrict__ x4 = (const float4*)x;

    float sq0 = 0.f, sq1 = 0.f, sq2 = 0.f, sq3 = 0.f;      // per-column sumsq
    float sm0 = 0.f, sm1 = 0.f, sm2 = 0.f, sm3 = 0.f;      // scalar sums (t<32)
    float acnt = 0.f;

    auto flush = [&](int g) {
        __syncthreads();                   // lsq safe to overwrite
        if (active) {
            lsq[4 * t + 0] = sq0;
            lsq[4 * t + 1] = sq1;
            lsq[4 * t + 2] = sq2;
            lsq[4 * t + 3] = sq3;
        }
        __syncthreads();
        for (int c = t; c < NCH; c += 128) {
            float s;
            if (c < 128) {
                s = lsq[c];
            } else if (c < 192) {
                int b = 128 + (c - 128) * 3;
                s = lsq[b] + lsq[b + 1] + lsq[b + 2];
            } else {
                int b = 320 + (c - 192) * 5;
                s = lsq[b] + lsq[b + 1] + lsq[b + 2] + lsq[b + 3] + lsq[b + 4];
            }
            atomicAdd(&ssq[g * NCH + c], s);
        }
        if (t < 32) {                      // scalar channels 4t..4t+3
            atomicAdd(&sums[g * 128 + 4 * t + 0], sm0);
            atomicAdd(&sums[g * 128 + 4 * t + 1], sm1);
            atomicAdd(&sums[g * 128 + 4 * t + 2], sm2);
            atomicAdd(&sums[g * 128 + 4 * t + 3], sm3);
        }
        if (t == 0) atomicAdd(&cnt[g], acnt);
    };

    int curg = seg[r0];
    for (int r = r0; r < r1; ++r) {
        int g = seg[r];                    // block-uniform
        if (g != curg) {                   // uniform branch: flush + reset
            flush(curg);
            curg = g;
            sq0 = sq1 = sq2 = sq3 = 0.f;
            sm0 = sm1 = sm2 = sm3 = 0.f;
            acnt = 0.f;
        }
        if (active) {
            if (r + 16 < r1)               // gfx1250: global_prefetch_b8
                __builtin_prefetch(&x4[(size_t)(r + 16) * DIM4 + t], 0, 0);
            float4 v = x4[(size_t)r * DIM4 + t];
            sq0 = fmaf(v.x, v.x, sq0);
            sq1 = fmaf(v.y, v.y, sq1);
            sq2 = fmaf(v.z, v.z, sq2);
            sq3 = fmaf(v.w, v.w, sq3);
            if (t < 32) { sm0 += v.x; sm1 += v.y; sm2 += v.z; sm3 += v.w; }
        }
        acnt += 1.0f;
    }
    flush(curg);
}

// ---------------- pass 2: fold stats into (scale, shift) tables --------------
__global__ void finalize_stats(const float* __restrict__ w,
                               const float* __restrict__ b,
                               float* __restrict__ ws)
{
    int i = blockIdx.x * blockDim.x + threadIdx.x;
    if (i >= NG * NCH) return;
    int g = i / NCH, c = i - g * NCH;

    const float* cnt  = ws + WS_CNT;
    const float* sums = ws + WS_SUM;
    const float* ssq  = ws + WS_SSQ;

    float n  = cnt[g];  if (n < 1.0f) n = 1.0f;
    float ss = ssq[g * NCH + c];
    float sc, sh = 0.0f;
    if (c < 128) {                       // scalar irrep: instance norm
        float m   = sums[g * 128 + c] / n;
        float var = ss / n - m * m;
        sc = 1.0f / sqrtf(var + NEPS) * w[c];
        sh = b[c] - m * sc;              // (x-m)*sc + b  ==  x*sc + sh
    } else {                             // l=1 (d=3) / l=2 (d=5)
        float d  = (c < 192) ? 3.0f : 5.0f;
        float ms = ss / (n * d);
        sc = 1.0f / sqrtf(ms + NEPS) * w[c];
    }
    ws[WS_SCALE + i] = sc;
    ws[WS_SHIFT + i] = sh;
}

// ---------------- pass 3: streaming normalize --------------------------------
// Double-buffered CDNA5 async DMA (global -> LDS, ASYNCcnt): issue tile k+1,
// s_wait_asynccnt 1 guarantees tile k landed (in-order completion), process,
// swap. Output stored non-temporal so it doesn't evict cached input from L2.
__global__ __launch_bounds__(256) void apply_norm(
    const float* __restrict__ x, const int* __restrict__ seg,
    const float* __restrict__ ws, float* __restrict__ out, int nrows)
{
    __shared__ float4 stage[2][256];
    const float* scale = ws + WS_SCALE;
    const float* shift = ws + WS_SHIFT;

    const size_t total4 = (size_t)nrows * DIM4;
    const size_t stride = (size_t)gridDim.x * blockDim.x;

    size_t i4 = (size_t)blockIdx.x * blockDim.x + threadIdx.x;
    if (i4 >= total4) return;            // no barriers below: early-exit safe

    const uint32_t lds0 = (uint32_t)(size_t)&stage[0][threadIdx.x];
    const uint32_t lds1 = (uint32_t)(size_t)&stage[1][threadIdx.x];

    // prime buffer 0
    asm volatile("global_load_async_to_lds_b128 %0, %1, off"
                 :: "v"(lds0),
                    "v"((unsigned long long)(size_t)(x + i4 * 4))
                 : "memory");

    int buf = 0;
    for (; i4 < total4; i4 += stride) {
        size_t nx  = i4 + stride;
        size_t nxc = (nx < total4) ? nx : i4;          // tail: dummy reload
        uint32_t dst = buf ? lds0 : lds1;
        asm volatile("global_load_async_to_lds_b128 %0, %1, off"
                     :: "v"(dst),
                        "v"((unsigned long long)(size_t)(x + nxc * 4))
                     : "memory");
        asm volatile("s_wait_asynccnt 0x1" ::: "memory");  // older tile landed

        float4 v = stage[buf][threadIdx.x];            // own slot only

        int row = (int)(i4 / DIM4);
        int c0  = (int)(i4 % DIM4) * 4;
        int g   = seg[row];
        const float* sc = scale + g * NCH;
        const float* sh = shift + g * NCH;

        float r[4] = {v.x, v.y, v.z, v.w};
#pragma unroll
        for (int k = 0; k < 4; ++k) {
            int col = c0 + k;
            int ch  = (col < 128) ? col
                    : (col < 320) ? 128 + (col - 128) / 3
                                  : 192 + (col - 320) / 5;
            r[k] = fmaf(r[k], sc[ch], sh[ch]);
        }
        v4f o = {r[0], r[1], r[2], r[3]};
        __builtin_nontemporal_store(o, (v4f*)out + i4);   // NT store
        buf ^= 1;
    }
}

extern "C" void kernel_launch(void* const* d_in, const int* in_sizes, int n_in,
                              void* d_out, int out_size, void* d_ws, size_t ws_size,
                              hipStream_t stream) {
    const float* x   = (const float*)d_in[0];
    const int*   seg = (const int*)d_in[1];
    const float* w   = (const float*)d_in[2];
    const float* b   = (const float*)d_in[3];
    float*       out = (float*)d_out;
    float*       ws  = (float*)d_ws;

    const int nrows = in_sizes[0] / DIM;

    zero_stats<<<(WS_ZERO_N + 255) / 256, 256, 0, stream>>>(ws, WS_ZERO_N);

    const int nb  = 2048;
    const int rpb = (nrows + nb - 1) / nb;
    pass1_stats<<<nb, 128, 0, stream>>>(x, seg, ws, nrows, rpb);

    finalize_stats<<<(NG * NCH + 255) / 256, 256, 0, stream>>>(w, b, ws);

    apply_norm<<<6144, 256, 0, stream>>>(x, seg, ws, out, nrows);
}